// Preprocessor_51634096833327
// MI455X (gfx1250) — compile-verified
//
#include <hip/hip_runtime.h>

typedef __attribute__((ext_vector_type(2))) float v2f;
typedef __attribute__((ext_vector_type(8))) float v8f;

#define HDIM 512
#define WDIM 512
#define BDIM 32
#define CDIM 4
#define HW   (HDIM * WDIM)

// GEMM blocking
#define MT 128          // block tile rows
#define NT 64           // block tile cols
#define KC 32           // K chunk staged in LDS
#define AS_STRIDE 34    // 32 + 2 pad: conflict-free strided column reads
#define BS_STRIDE 66    // 64 + 2 pad: rows stay 8B-aligned

// ---------------------------------------------------------------------------
// 1) Gaussian matrix K[i][j] = exp(-(i-j)^2 / 2), sigma = 1 (symmetric: both
//    Kx and Ky).
// ---------------------------------------------------------------------------
__global__ void build_gauss(float* __restrict__ K) {
    int idx = blockIdx.x * blockDim.x + threadIdx.x;      // [0, 512*512)
    int i = idx >> 9;
    int j = idx & 511;
    float d = (float)(i - j);
    K[idx] = __expf(-0.5f * d * d);
}

// ---------------------------------------------------------------------------
// 2) Zero per-batch max accumulators (d_ws is poisoned by the harness).
// ---------------------------------------------------------------------------
__global__ void zero_max(unsigned int* __restrict__ gmax) {
    if (threadIdx.x < BDIM) gmax[threadIdx.x] = 0u;
}

// ---------------------------------------------------------------------------
// LDS-blocked WMMA GEMM:  D[b] = op(A[b]) x B[b]   (512x512x512 f32)
//   - 256 threads / 8 waves per block; block computes a 128x64 output tile.
//   - Wave w computes rows [w*16, w*16+16) x all 64 cols -> 4 independent
//     16x16 accumulators fed by V_WMMA_F32_16X16X4_F32 (A fragment reused 4x).
//   - BINARIZE: apply (x > 0) ? 1 : 0 while filling the A tile (GEMM1).
//   - DOMAX:    fold per-batch max of D via wave32 shfl_xor + bit atomicMax
//               (values are >= 0, so uint bit-compare is order-preserving).
// WMMA fragment layouts per ISA 7.12.2:
//   A 16x4: lanes 0-15 hold M=lane with K={0,1} in {v0,v1}; lanes 16-31 K={2,3}.
//   B 4x16: v0 = row K (lanes 0-15) / row K+2 (lanes 16-31); v1 = row +1.
//   C/D:    vgpr v holds C[M = v + 8*(lane>=16)][N = lane%16].
// ---------------------------------------------------------------------------
template <bool BINARIZE, bool DOMAX>
__global__ void gemm_tile(const float* __restrict__ Amat, size_t a_bs,
                          const float* __restrict__ Bmat, size_t b_bs,
                          float* __restrict__ Dmat,       size_t d_bs,
                          unsigned int* __restrict__ gmax) {
    __shared__ float As[MT * AS_STRIDE];   // 17408 B
    __shared__ float Bs[KC * BS_STRIDE];   //  8448 B

    const int t    = threadIdx.x;
    const int lane = t & 31;
    const int wv   = t >> 5;               // wave id 0..7
    const int lm   = lane & 15;
    const int hi   = lane >> 4;            // 0 or 1 -> K offset {0,1} vs {2,3}

    const int b   = blockIdx.z;
    const int tn0 = blockIdx.x * NT;       // 8 N-tiles
    const int tm0 = blockIdx.y * MT;       // 4 M-tiles

    const float* A = Amat + (size_t)b * a_bs;
    const float* B = Bmat + (size_t)b * b_bs;
    float*       D = Dmat + (size_t)b * d_bs;

    // LDS fill assignments (uniform trip counts -> EXEC stays all-ones)
    const int aseg  = t & 7;               // 8 x float4 segments per 32-wide row
    const int arow0 = t >> 3;              // 0..31, step 32 over 128 rows
    const int bseg  = t & 15;              // 16 x float4 segments per 64-wide row
    const int brow0 = t >> 4;              // 0..15, step 16 over 32 rows

    v8f c[4] = {{}, {}, {}, {}};

    for (int kc = 0; kc < WDIM; kc += KC) {
        // ---- stage A tile: rows [tm0, tm0+128) x cols [kc, kc+32) ----
        #pragma unroll
        for (int r = 0; r < MT; r += 32) {
            const int row = arow0 + r;
            float4 v = *(const float4*)(A + (size_t)(tm0 + row) * WDIM + kc + aseg * 4);
            if (BINARIZE) {
                v.x = (v.x > 0.0f) ? 1.0f : 0.0f;
                v.y = (v.y > 0.0f) ? 1.0f : 0.0f;
                v.z = (v.z > 0.0f) ? 1.0f : 0.0f;
                v.w = (v.w > 0.0f) ? 1.0f : 0.0f;
            }
            float* dst = &As[row * AS_STRIDE + aseg * 4];
            dst[0] = v.x; dst[1] = v.y; dst[2] = v.z; dst[3] = v.w;
        }
        // ---- stage B tile: rows [kc, kc+32) x cols [tn0, tn0+64) ----
        #pragma unroll
        for (int r = 0; r < KC; r += 16) {
            const int row = brow0 + r;
            float4 v = *(const float4*)(B + (size_t)(kc + row) * WDIM + tn0 + bseg * 4);
            float* dst = &Bs[row * BS_STRIDE + bseg * 4];
            dst[0] = v.x; dst[1] = v.y; dst[2] = v.z; dst[3] = v.w;
        }
        __syncthreads();

        // ---- 8 K-steps of 4, 4 N-accumulators each ----
        #pragma unroll
        for (int k = 0; k < KC; k += 4) {
            const int ka = k + 2 * hi;
            v2f a;
            a.x = As[(wv * 16 + lm) * AS_STRIDE + ka];
            a.y = As[(wv * 16 + lm) * AS_STRIDE + ka + 1];
            #pragma unroll
            for (int n = 0; n < 4; ++n) {
                v2f bb;
                bb.x = Bs[ka * BS_STRIDE + n * 16 + lm];
                bb.y = Bs[(ka + 1) * BS_STRIDE + n * 16 + lm];
                c[n] = __builtin_amdgcn_wmma_f32_16x16x4_f32(
                           false, a, false, bb, (short)0, c[n], false, false);
            }
        }
        __syncthreads();
    }

    // ---- write out 16x64 strip per wave ----
    float mx = 0.0f;
    #pragma unroll
    for (int n = 0; n < 4; ++n) {
        #pragma unroll
        for (int v = 0; v < 8; ++v) {
            const int m = tm0 + wv * 16 + v + 8 * hi;
            D[(size_t)m * WDIM + tn0 + n * 16 + lm] = c[n][v];
            if (DOMAX) mx = fmaxf(mx, c[n][v]);
        }
    }
    if (DOMAX) {
        #pragma unroll
        for (int off = 16; off > 0; off >>= 1)
            mx = fmaxf(mx, __shfl_xor(mx, off, 32));
        if (lane == 0)
            atomicMax(&gmax[b], __float_as_uint(mx));
    }
}

// ---------------------------------------------------------------------------
// 5) Copy channels 0,1,3 through; scale channel 2 in place by 1/max (0 -> 1).
// ---------------------------------------------------------------------------
__global__ void finalize(const float* __restrict__ x,
                         float* __restrict__ out,
                         const unsigned int* __restrict__ gmax) {
    size_t idx = (size_t)blockIdx.x * blockDim.x + threadIdx.x;
    size_t b  = idx / ((size_t)CDIM * HW);
    size_t r  = idx % ((size_t)CDIM * HW);
    size_t ch = r / HW;
    if (ch == 2) {
        float m = __uint_as_float(gmax[b]);
        if (m == 0.0f) m = 1.0f;
        out[idx] = out[idx] * (1.0f / m);
    } else {
        out[idx] = x[idx];
    }
}

// ---------------------------------------------------------------------------
// Launch. Workspace layout: [K 1MiB][T 32MiB][gmax 128B]  => ~33 MiB used.
// ---------------------------------------------------------------------------
extern "C" void kernel_launch(void* const* d_in, const int* in_sizes, int n_in,
                              void* d_out, int out_size, void* d_ws, size_t ws_size,
                              hipStream_t stream) {
    const float* x   = (const float*)d_in[0];
    float*       out = (float*)d_out;
    char*        ws  = (char*)d_ws;

    float*        Kmat = (float*)ws;
    float*        T    = (float*)(ws + ((size_t)1 << 20));
    unsigned int* gmax = (unsigned int*)(ws + ((size_t)33 << 20));

    build_gauss<<<HW / 256, 256, 0, stream>>>(Kmat);
    zero_max<<<1, 32, 0, stream>>>(gmax);

    dim3 gemm_grid(WDIM / NT, HDIM / MT, BDIM);   // (8, 4, 32)

    // GEMM1: T[b] = binarize(x[b, ch2]) x K
    gemm_tile<true, false><<<gemm_grid, 256, 0, stream>>>(
        x + (size_t)2 * HW, (size_t)CDIM * HW,   // A: channel 2, batch stride 4*HW
        Kmat, 0,                                  // B: shared Gaussian matrix
        T, (size_t)HW,                            // D: T
        nullptr);

    // GEMM2: out[b, ch2] = K x T[b]  (+ per-batch max)
    gemm_tile<false, true><<<gemm_grid, 256, 0, stream>>>(
        Kmat, 0,                                  // A: Gaussian matrix
        T, (size_t)HW,                            // B: T
        out + (size_t)2 * HW, (size_t)CDIM * HW,  // D: channel 2 of output
        gmax);

    const size_t total = (size_t)BDIM * CDIM * HW;   // 33,554,432
    finalize<<<(unsigned)(total / 256), 256, 0, stream>>>(x, out, gmax);
}